// warpImage_pyTorchBicubic_86423331930421
// MI455X (gfx1250) — compile-verified
//
#include <hip/hip_runtime.h>
#include <hip/hip_bf16.h>
#include <stdint.h>

typedef __attribute__((ext_vector_type(2))) float v2f;
typedef __attribute__((ext_vector_type(8))) float v8f;

#define IMGD 28
#define CRS  7
#define IPB  8                       // images per block (8 waves, 1 image/wave)
#define IMGF (IMGD*IMGD)             // 784
#define FLWF (CRS*CRS*2)             // 98

// LDS layout in float units (all region starts 16B aligned)
#define OFF_IMG  0                   // IPB*784          = 6272
#define OFF_FRAW 6272                // IPB*98           = 784  (async landing pad)
#define OFF_W    7056                // 32*8             = 256  (zero-padded 28x7)
#define OFF_FP   7312                // IPB*8*16*2       = 2048 (zero-padded flow)
#define OFF_T    9360                // IPB*2*32*16      = 8192 (padded T tiles)
#define OFF_V    17552               // IPB*32*32*2      = 16384 (padded V tiles)
#define LDS_FLOATS 33936             // 135,744 bytes

__device__ __forceinline__ void cubw(float t, float* w) {
  float t2 = t*t, t3 = t2*t;
  w[0] = -0.5f*t3 +      t2 - 0.5f*t;
  w[1] =  1.5f*t3 - 2.5f*t2 + 1.0f;
  w[2] = -1.5f*t3 + 2.0f*t2 + 0.5f*t;
  w[3] =  0.5f*t3 - 0.5f*t2;
}

__device__ __forceinline__ unsigned lds_off(const void* p) {
  return (unsigned)(uintptr_t)p;     // flat LDS address: low 32 bits = LDS byte offset
}

// CDNA5 async global->LDS copy (ASYNCcnt path), GVS addressing: saddr + voffset
__device__ __forceinline__ void async_b128(unsigned lds, const float* base, unsigned byteOff) {
  asm volatile("global_load_async_to_lds_b128 %0, %1, %2"
               :: "v"(lds), "v"(byteOff), "s"(base) : "memory");
}

extern "C" __global__ __launch_bounds__(256)
void bicubic_warp_kernel(const float* __restrict__ x,
                         const float* __restrict__ y,
                         float* __restrict__ out) {
  extern __shared__ float smem[];
  const int tid  = threadIdx.x;
  const int lane = tid & 31;
  const int wv   = tid >> 5;         // wave id == image-in-block id
  const int half = lane >> 4;        // 0: lanes 0-15, 1: lanes 16-31
  const int lm   = lane & 15;
  const int b0   = blockIdx.x * IPB;

  // ---------------- async copy: 8 images + 8 flows into LDS ----------------
  {
    const float* gimg = x + (size_t)b0 * IMGF;   // 25088 B contiguous, 16B aligned
    const float* gflw = y + (size_t)b0 * FLWF;   // 3136 B contiguous, 16B aligned
    unsigned limg = lds_off(&smem[OFF_IMG]);
    unsigned lflw = lds_off(&smem[OFF_FRAW]);
    for (int i = tid; i < (IPB*IMGF)/4; i += 256)      // 1568 x b128
      async_b128(limg + (unsigned)i*16u, gimg, (unsigned)i*16u);
    if (tid < (IPB*FLWF)/4)                            // 196 x b128
      async_b128(lflw + (unsigned)tid*16u, gflw, (unsigned)tid*16u);
  }

  // ---- build zero-padded 32x8 interpolation matrix W while DMA runs -------
  {
    const int o = tid >> 3, k = tid & 7;               // tid < 256 covers 32x8
    float val = 0.0f;
    if (o < IMGD && k < CRS) {
      float src = ((float)(o*6)) / 27.0f;              // arange*(in-1)/(out-1)
      float i0f = floorf(src);
      float t   = src - i0f;
      int   i0  = (int)i0f;
      float w4[4]; cubw(t, w4);
      #pragma unroll
      for (int d = 0; d < 4; ++d) {
        int r = i0 - 1 + d;
        r = r < 0 ? 0 : (r > CRS-1 ? CRS-1 : r);       // clamped taps accumulate
        if (r == k) val += w4[d];
      }
    }
    smem[OFF_W + tid] = val;
    #pragma unroll
    for (int i = 0; i < (IPB*256)/256; ++i)            // zero padded flow region
      smem[OFF_FP + tid + i*256] = 0.0f;
  }

  asm volatile("s_wait_asynccnt 0x0" ::: "memory");
  __syncthreads();

  // ---- repack this wave's flow into zero-padded [8][16][2] tile -----------
  {
    const float* raw = &smem[OFF_FRAW + wv*FLWF];
    float*       fp  = &smem[OFF_FP   + wv*256];
    for (int i = lane; i < FLWF; i += 32) {
      int k = i / 14, r = i - k*14;
      int n = r >> 1, c = r & 1;
      fp[(k*16 + n)*2 + c] = raw[i];
    }
  }
  __syncthreads();

  // ---------------- stage 1: T = W (28x7) @ F (7x7) per channel ------------
  // WMMA f32 16x16x4; A: m=lm, k=v+2*half ; B: n=lm, k=v+2*half ; D: m=v+8*half
  const float* Wp = &smem[OFF_W];
  const float* Fp = &smem[OFF_FP + wv*256];
  float*       Tl = &smem[OFF_T  + wv*(2*32*16)];
  #pragma unroll
  for (int c = 0; c < 2; ++c) {
    #pragma unroll
    for (int mt = 0; mt < 2; ++mt) {
      const int mbase = mt*16;
      v8f acc = {};
      #pragma unroll
      for (int kb = 0; kb < 2; ++kb) {
        const int k0 = kb*4 + 2*half;
        const int o  = mbase + lm;
        v2f a, b;
        a.x = Wp[o*8 + k0];                            // zero-padded, unconditional
        a.y = Wp[o*8 + k0 + 1];
        b.x = Fp[(k0*16 + lm)*2 + c];
        b.y = Fp[((k0+1)*16 + lm)*2 + c];
        acc = __builtin_amdgcn_wmma_f32_16x16x4_f32(false, a, false, b,
                                                    (short)0, acc, false, false);
      }
      #pragma unroll
      for (int v = 0; v < 8; ++v) {                    // full-tile store, no guards
        const int o = mbase + v + 8*half;
        Tl[c*512 + o*16 + lm] = acc[v];
      }
    }
  }

  __syncthreads();

  // ---------------- stage 2: V = T (28x7) @ W^T (7x28) per channel ---------
  float* Vl = &smem[OFF_V + wv*(32*32*2)];
  #pragma unroll
  for (int c = 0; c < 2; ++c) {
    #pragma unroll
    for (int mt = 0; mt < 2; ++mt) {
      #pragma unroll
      for (int nt = 0; nt < 2; ++nt) {
        const int mbase = mt*16, nbase = nt*16;
        v8f acc = {};
        #pragma unroll
        for (int kb = 0; kb < 2; ++kb) {
          const int k0 = kb*4 + 2*half;
          const int o  = mbase + lm;
          const int n  = nbase + lm;
          v2f a, b;
          a.x = Tl[c*512 + o*16 + k0];                 // padded, unconditional
          a.y = Tl[c*512 + o*16 + k0 + 1];
          b.x = Wp[n*8 + k0];                          // B[k,n] = W[n,k]
          b.y = Wp[n*8 + k0 + 1];
          acc = __builtin_amdgcn_wmma_f32_16x16x4_f32(false, a, false, b,
                                                      (short)0, acc, false, false);
        }
        #pragma unroll
        for (int v = 0; v < 8; ++v) {                  // full-tile store, no guards
          const int o = mbase + v + 8*half;
          const int n = nbase + lm;
          Vl[(o*32 + n)*2 + c] = acc[v];
        }
      }
    }
  }

  __syncthreads();

  // ---------------- warp phase: 16-tap bicubic gather from LDS -------------
  const float* img  = &smem[OFF_IMG + wv*IMGF];
  float*       outw = out + (size_t)(b0 + wv) * IMGF;
  int o  = lane / IMGD;                                // incremental row/col
  int wc = lane - o*IMGD;
  for (int p = lane; p < IMGF; p += 32) {
    const float vx = Vl[(o*32 + wc)*2 + 0];
    const float vy = Vl[(o*32 + wc)*2 + 1];
    const float sx = (float)wc + vx;
    const float sy = (float)o  + vy;
    const float x0f = floorf(sx), y0f = floorf(sy);
    const int   x0  = (int)x0f,   y0  = (int)y0f;
    float wx[4], wy4[4];
    cubw(sx - x0f, wx);
    cubw(sy - y0f, wy4);
    const float* rows[4];
    int xs[4];
    #pragma unroll
    for (int i = 0; i < 4; ++i) {
      int yi = y0 + i - 1;
      yi = yi < 0 ? 0 : (yi > IMGD-1 ? IMGD-1 : yi);
      rows[i] = img + yi*IMGD;
    }
    #pragma unroll
    for (int j = 0; j < 4; ++j) {
      int xj = x0 + j - 1;
      xs[j] = xj < 0 ? 0 : (xj > IMGD-1 ? IMGD-1 : xj);
    }
    float acc = 0.0f;
    #pragma unroll
    for (int i = 0; i < 4; ++i) {
      float r = wx[0]*rows[i][xs[0]] + wx[1]*rows[i][xs[1]]
              + wx[2]*rows[i][xs[2]] + wx[3]*rows[i][xs[3]];
      acc = fmaf(wy4[i], r, acc);
    }
    outw[p] = acc;

    wc += 32 - IMGD; ++o;                              // p += 32 => col+4, row+1
    if (wc >= IMGD) { wc -= IMGD; ++o; }
  }
}

extern "C" void kernel_launch(void* const* d_in, const int* in_sizes, int n_in,
                              void* d_out, int out_size, void* d_ws, size_t ws_size,
                              hipStream_t stream) {
  const float* x = (const float*)d_in[0];   // (B, 28, 28) f32
  const float* y = (const float*)d_in[1];   // (B, 7, 7, 2) f32
  float* out = (float*)d_out;               // (B, 28, 28) f32
  const int Bn = in_sizes[0] / IMGF;        // 32768
  const int blocks = Bn / IPB;              // 4096
  const size_t shmem = (size_t)LDS_FLOATS * sizeof(float);
  bicubic_warp_kernel<<<blocks, 256, shmem, stream>>>(x, y, out);
}